// ConditionalRandomField_7086696038873
// MI455X (gfx1250) — compile-verified
//
#include <hip/hip_runtime.h>
#include <hip/hip_bf16.h>

#define THRESH -1000000.0f

typedef float v2f __attribute__((ext_vector_type(2)));
typedef float v8f __attribute__((ext_vector_type(8)));

constexpr int B  = 512;
constexpr int T  = 512;
constexpr int KS = 8;
constexpr int KT = 64;

// block roles in the fused launch (blockDim = 32 everywhere)
constexpr int NB_TAG = B / 16;   // 32  : tag NLL, one wave per 16 rows (WMMA)
constexpr int NB_SEQ = B / 32;   // 16  : seq NLL, one row per lane
constexpr int NB_VT  = B;        // 512 : tag viterbi, 2 tags per lane
constexpr int NB_VS  = B / 4;    // 128 : seq viterbi, 4 sequences per block

constexpr int SMEM_BYTES = 49664;  // max over roles (viterbi64: 49600)

// CDNA5 async global->LDS (ASYNCcnt) -----------------------------------------
#define ASYNC_LOAD_B128(ldsoff, gptr)                                   \
    asm volatile("global_load_async_to_lds_b128 %0, %1, off"            \
                 :: "v"(ldsoff), "v"(gptr) : "memory")
#define WAIT_ASYNC0() asm volatile("s_wait_asynccnt 0x0" ::: "memory")

// ---------------------------------------------------------------------------
// Tag-CRF forward NLL, K=64, V_WMMA_F32_16X16X4_F32, async-pipelined emissions.
// One wave per 16 batch rows.
// newalpha = Mrow + log( exp(alpha - Mrow) @ exp(trans) ) + emit, masked.
// ---------------------------------------------------------------------------
__device__ __forceinline__ void tag_nll_body(
    const float* __restrict__ emis,   // [B,T,64]
    const int*   __restrict__ ytrue,  // [B,T]
    const float* __restrict__ trans,  // [64,64]
    float* __restrict__ nll_out,      // [B]
    int b0, char* smem)
{
    float* E     = (float*)smem;      // exp(trans): 4096 f
    float* alpha = E + 4096;          // 16*68 padded
    float* emitA = alpha + 1088;      // double-buffered emissions
    float* emitB = emitA + 1088;
    float* rmax  = emitB + 1088;      // 16*2
    float* mpart = rmax + 32;         // 16*2
    float* mrow  = mpart + 32;        // 16

    const int lane  = threadIdx.x;    // 0..31
    const int l15   = lane & 15;
    const int h16   = lane >> 4;
    const int em    = lane >> 1;      // emission row 0..15
    const int ehalf = lane & 1;       // half 0/1

    for (int i = lane; i < 64 * 64; i += 32) E[i] = __expf(trans[i]);

    const float* myrow = emis + ((size_t)(b0 + em) * T) * 64 + ehalf * 32;
    const unsigned ldsA = (unsigned)(size_t)(void*)&emitA[em * 68 + ehalf * 32];
    const unsigned ldsB = (unsigned)(size_t)(void*)&emitB[em * 68 + ehalf * 32];

    // ---- t = 0 : async into A, then init ----
#pragma unroll
    for (int q = 0; q < 8; ++q) ASYNC_LOAD_B128(ldsA + 16u * q, myrow + 4 * q);
    WAIT_ASYNC0();
    __syncthreads();
    {
        const float4* s = (const float4*)&emitA[em * 68 + ehalf * 32];
        float4 v[8];
        bool ok = true;
#pragma unroll
        for (int q = 0; q < 8; ++q) {
            v[q] = s[q];
            ok = ok && (v[q].x > THRESH) && (v[q].y > THRESH) &&
                       (v[q].z > THRESH) && (v[q].w > THRESH);
        }
        mpart[em * 2 + ehalf] = ok ? 1.0f : 0.0f;
        __syncthreads();
        float m0 = mpart[em * 2 + 0] * mpart[em * 2 + 1];
        float4* ad = (float4*)&alpha[em * 68 + ehalf * 32];
#pragma unroll
        for (int q = 0; q < 8; ++q) {
            float4 w = v[q];
            w.x *= m0; w.y *= m0; w.z *= m0; w.w *= m0;
            ad[q] = w;
        }
    }
    float point = 0.f, transs = 0.f, mprev = 0.f;
    int yprev = 0;
    if (lane < 16) {
        mprev = mpart[lane * 2 + 0] * mpart[lane * 2 + 1];
        yprev = ytrue[(size_t)(b0 + lane) * T + 0];
        point = mprev * emitA[lane * 68 + yprev];
    }
    // kick off t = 1 into B
#pragma unroll
    for (int q = 0; q < 8; ++q) ASYNC_LOAD_B128(ldsB + 16u * q, myrow + 64 + 4 * q);

    // ---- serial scan ----
    for (int t = 1; t < T; ++t) {
        WAIT_ASYNC0();          // buf[t&1] ready
        __syncthreads();
        float* eb = (t & 1) ? emitB : emitA;
        // overlap next step's DMA with this step's compute
        if (t + 1 < T) {
            const unsigned ldsN = (t & 1) ? ldsA : ldsB;
            const float* nrow = myrow + (size_t)(t + 1) * 64;
#pragma unroll
            for (int q = 0; q < 8; ++q) ASYNC_LOAD_B128(ldsN + 16u * q, nrow + 4 * q);
        }
        // validity-mask partials for step t
        {
            const float4* s = (const float4*)&eb[em * 68 + ehalf * 32];
            bool ok = true;
#pragma unroll
            for (int q = 0; q < 8; ++q) {
                float4 v = s[q];
                ok = ok && (v.x > THRESH) && (v.y > THRESH) &&
                           (v.z > THRESH) && (v.w > THRESH);
            }
            mpart[em * 2 + ehalf] = ok ? 1.0f : 0.0f;
        }
        // row-max of alpha (state t-1), half-row per lane
        {
            const float4* ar = (const float4*)&alpha[em * 68 + ehalf * 32];
            float mx = -__builtin_inff();
#pragma unroll
            for (int q = 0; q < 8; ++q) {
                float4 v = ar[q];
                mx = fmaxf(mx, fmaxf(fmaxf(v.x, v.y), fmaxf(v.z, v.w)));
            }
            rmax[em * 2 + ehalf] = mx;
        }
        __syncthreads();
        // gold-path scores (lanes < 16; trans table is L2-hot)
        if (lane < 16) {
            float mk = mpart[lane * 2 + 0] * mpart[lane * 2 + 1];
            mrow[lane] = mk;
            int yt = ytrue[(size_t)(b0 + lane) * T + t];
            point  += mk * eb[lane * 68 + yt];
            transs += mprev * mk * trans[yprev * 64 + yt];
            yprev = yt; mprev = mk;
        }
        __syncthreads();

        // P = exp(alpha - Mrow); D = P @ E via f32 WMMA 16x16x4
        float Ml = fmaxf(rmax[l15 * 2 + 0], rmax[l15 * 2 + 1]);
        v8f acc0 = {0,0,0,0,0,0,0,0};
        v8f acc1 = {0,0,0,0,0,0,0,0};
        v8f acc2 = {0,0,0,0,0,0,0,0};
        v8f acc3 = {0,0,0,0,0,0,0,0};
        for (int k0 = 0; k0 < 64; k0 += 4) {
            const int kb = k0 + 2 * h16;
            v2f a;
            a.x = __expf(alpha[l15 * 68 + kb + 0] - Ml);
            a.y = __expf(alpha[l15 * 68 + kb + 1] - Ml);
            v2f bv0, bv1, bv2, bv3;
            bv0.x = E[kb * 64 +  0 + l15]; bv0.y = E[(kb + 1) * 64 +  0 + l15];
            bv1.x = E[kb * 64 + 16 + l15]; bv1.y = E[(kb + 1) * 64 + 16 + l15];
            bv2.x = E[kb * 64 + 32 + l15]; bv2.y = E[(kb + 1) * 64 + 32 + l15];
            bv3.x = E[kb * 64 + 48 + l15]; bv3.y = E[(kb + 1) * 64 + 48 + l15];
            acc0 = __builtin_amdgcn_wmma_f32_16x16x4_f32(false, a, false, bv0, (short)0, acc0, false, false);
            acc1 = __builtin_amdgcn_wmma_f32_16x16x4_f32(false, a, false, bv1, (short)0, acc1, false, false);
            acc2 = __builtin_amdgcn_wmma_f32_16x16x4_f32(false, a, false, bv2, (short)0, acc2, false, false);
            acc3 = __builtin_amdgcn_wmma_f32_16x16x4_f32(false, a, false, bv3, (short)0, acc3, false, false);
        }
        // newalpha = Mrow + log(D) + emit, masked blend with old alpha
        {
            const int m2b = h16 * 8;
#pragma unroll
            for (int r = 0; r < 8; ++r) {
                const int m2   = m2b + r;
                const float Mm = fmaxf(rmax[m2 * 2 + 0], rmax[m2 * 2 + 1]);
                const float mk = mrow[m2];
                {
                    int n = 0 + l15;
                    float old = alpha[m2 * 68 + n];
                    float v = Mm + __logf(acc0[r]) + eb[m2 * 68 + n];
                    alpha[m2 * 68 + n] = mk * v + (1.f - mk) * old;
                }
                {
                    int n = 16 + l15;
                    float old = alpha[m2 * 68 + n];
                    float v = Mm + __logf(acc1[r]) + eb[m2 * 68 + n];
                    alpha[m2 * 68 + n] = mk * v + (1.f - mk) * old;
                }
                {
                    int n = 32 + l15;
                    float old = alpha[m2 * 68 + n];
                    float v = Mm + __logf(acc2[r]) + eb[m2 * 68 + n];
                    alpha[m2 * 68 + n] = mk * v + (1.f - mk) * old;
                }
                {
                    int n = 48 + l15;
                    float old = alpha[m2 * 68 + n];
                    float v = Mm + __logf(acc3[r]) + eb[m2 * 68 + n];
                    alpha[m2 * 68 + n] = mk * v + (1.f - mk) * old;
                }
            }
        }
    }

    __syncthreads();
    if (lane < 16) {
        float mx = -__builtin_inff();
        for (int k = 0; k < 64; ++k) mx = fmaxf(mx, alpha[lane * 68 + k]);
        float sm = 0.f;
        for (int k = 0; k < 64; ++k) sm += __expf(alpha[lane * 68 + k] - mx);
        nll_out[b0 + lane] = mx + __logf(sm) - (point + transs);
    }
}

// ---------------------------------------------------------------------------
// Seq-CRF forward NLL, K=8: one batch row per lane, alpha in registers.
// ---------------------------------------------------------------------------
__device__ __forceinline__ void seq_nll_body(
    const float* __restrict__ emis, const int* __restrict__ ytrue,
    const float* __restrict__ trans, float* __restrict__ nll_out,
    int b0, char* smem)
{
    float* tr = (float*)smem;  // 64 floats
    const int lane = threadIdx.x;
    for (int i = lane; i < KS * KS; i += 32) tr[i] = trans[i];
    __syncthreads();

    const int b = b0 + lane;
    const float* row = emis + (size_t)b * T * KS;
    const int*   yr  = ytrue + (size_t)b * T;

    float a[KS], e[KS];
    {
        const float4* s = (const float4*)row;
        float4 v0 = s[0], v1 = s[1];
        e[0]=v0.x; e[1]=v0.y; e[2]=v0.z; e[3]=v0.w;
        e[4]=v1.x; e[5]=v1.y; e[6]=v1.z; e[7]=v1.w;
    }
    bool ok0 = true;
#pragma unroll
    for (int j = 0; j < KS; ++j) ok0 = ok0 && (e[j] > THRESH);
    float m0 = ok0 ? 1.0f : 0.0f;
#pragma unroll
    for (int j = 0; j < KS; ++j) a[j] = e[j] * m0;

    int   yprev = yr[0];
    float ey = 0.f;
#pragma unroll
    for (int j = 0; j < KS; ++j) ey = (j == yprev) ? e[j] : ey;
    float point = m0 * ey, transs = 0.f, mprev = m0;

    for (int t = 1; t < T; ++t) {
        const float4* s = (const float4*)(row + (size_t)t * KS);
        float4 v0 = s[0], v1 = s[1];
        e[0]=v0.x; e[1]=v0.y; e[2]=v0.z; e[3]=v0.w;
        e[4]=v1.x; e[5]=v1.y; e[6]=v1.z; e[7]=v1.w;
        if (t + 1 < T) __builtin_prefetch(row + (size_t)(t + 1) * KS, 0, 1);
        bool ok = true;
#pragma unroll
        for (int j = 0; j < KS; ++j) ok = ok && (e[j] > THRESH);
        float mk = ok ? 1.0f : 0.0f;

        int yt = yr[t];
        float eyt = 0.f;
#pragma unroll
        for (int j = 0; j < KS; ++j) eyt = (j == yt) ? e[j] : eyt;
        point  += mk * eyt;
        transs += mprev * mk * tr[yprev * KS + yt];
        yprev = yt; mprev = mk;

        float na[KS];
#pragma unroll
        for (int j = 0; j < KS; ++j) {
            float mx = -__builtin_inff();
#pragma unroll
            for (int i = 0; i < KS; ++i) mx = fmaxf(mx, a[i] + tr[i * KS + j]);
            float sm = 0.f;
#pragma unroll
            for (int i = 0; i < KS; ++i) sm += __expf(a[i] + tr[i * KS + j] - mx);
            float v = mx + __logf(sm) + e[j];
            na[j] = mk * v + (1.f - mk) * a[j];
        }
#pragma unroll
        for (int j = 0; j < KS; ++j) a[j] = na[j];
    }

    float mx = -__builtin_inff();
#pragma unroll
    for (int j = 0; j < KS; ++j) mx = fmaxf(mx, a[j]);
    float sm = 0.f;
#pragma unroll
    for (int j = 0; j < KS; ++j) sm += __expf(a[j] - mx);
    nll_out[b] = mx + __logf(sm) - (point + transs);
}

// ---------------------------------------------------------------------------
// Tag Viterbi, K=64: one sequence per block, 2 tags per lane, bp in LDS.
// ---------------------------------------------------------------------------
__device__ __forceinline__ void viterbi64_body(
    const float* __restrict__ emis, const float* __restrict__ trans,
    float* __restrict__ path_out, int b, char* smem)
{
    float* tr = (float*)smem;                         // 4096 f
    float* al = tr + 4096;                            // 2 x 64
    unsigned char* bp = (unsigned char*)(al + 128);   // 511 x 64

    const int lane = threadIdx.x;
    const int j0 = lane, j1 = lane + 32;
    for (int i = lane; i < 64 * 64; i += 32) tr[i] = trans[i];
    const float* row = emis + (size_t)b * T * 64;
    al[j0] = row[j0];
    al[j1] = row[j1];
    __syncthreads();

    for (int t = 1; t < T; ++t) {
        const int cur = (t & 1) * 64, prv = 64 - cur;
        const float e0 = row[(size_t)t * 64 + j0];
        const float e1 = row[(size_t)t * 64 + j1];
        if (t + 1 < T) __builtin_prefetch(row + (size_t)(t + 1) * 64 + j0, 0, 1);
        float b0v = -__builtin_inff(), b1v = -__builtin_inff();
        int i0 = 0, i1 = 0;
        for (int i = 0; i < 64; ++i) {
            const float ai = al[prv + i];
            const float s0 = ai + tr[i * 64 + j0];
            const float s1 = ai + tr[i * 64 + j1];
            if (s0 > b0v) { b0v = s0; i0 = i; }   // strict '>' => first-argmax
            if (s1 > b1v) { b1v = s1; i1 = i; }
        }
        al[cur + j0] = b0v + e0;
        al[cur + j1] = b1v + e1;
        bp[(t - 1) * 64 + j0] = (unsigned char)i0;
        bp[(t - 1) * 64 + j1] = (unsigned char)i1;
        __syncthreads();
    }

    if (lane == 0) {
        const int fin = ((T - 1) & 1) * 64;
        int last = 0; float bv = al[fin];
        for (int k = 1; k < 64; ++k) {
            float v = al[fin + k];
            if (v > bv) { bv = v; last = k; }
        }
        path_out[(size_t)b * T + (T - 1)] = (float)last;
        for (int t = T - 2; t >= 0; --t) {
            last = bp[t * 64 + last];
            path_out[(size_t)b * T + t] = (float)last;
        }
    }
}

// ---------------------------------------------------------------------------
// Seq Viterbi, K=8: 4 sequences per block (lane -> seq = lane>>3, j = lane&7).
// ---------------------------------------------------------------------------
__device__ __forceinline__ void viterbi8_body(
    const float* __restrict__ emis, const float* __restrict__ trans,
    float* __restrict__ path_out, int b0, char* smem)
{
    float* tr = (float*)smem;                        // 64 f
    float* al = tr + 64;                             // 4 seq x 2 x 8
    unsigned char* bp = (unsigned char*)(al + 64);   // 4 x 511 x 8

    const int lane = threadIdx.x;
    const int s = lane >> 3;          // sequence 0..3
    const int j = lane & 7;
    for (int i = lane; i < KS * KS; i += 32) tr[i] = trans[i];
    const float* row = emis + (size_t)(b0 + s) * T * KS;
    float* als = al + s * 16;
    unsigned char* bps = bp + s * (T - 1) * KS;
    als[j] = row[j];
    __syncthreads();

    for (int t = 1; t < T; ++t) {
        const int cur = (t & 1) * 8, prv = 8 - cur;
        const float e = row[(size_t)t * KS + j];
        if (t + 1 < T) __builtin_prefetch(row + (size_t)(t + 1) * KS + j, 0, 1);
        float bb = -__builtin_inff(); int bi = 0;
#pragma unroll
        for (int i = 0; i < KS; ++i) {
            float sc = als[prv + i] + tr[i * KS + j];
            if (sc > bb) { bb = sc; bi = i; }
        }
        als[cur + j] = bb + e;
        bps[(t - 1) * KS + j] = (unsigned char)bi;
        __syncthreads();
    }

    if (j == 0) {  // lanes 0,8,16,24: one per sequence
        const int fin = ((T - 1) & 1) * 8;
        int last = 0; float bv = als[fin];
#pragma unroll
        for (int k = 1; k < KS; ++k) {
            float v = als[fin + k];
            if (v > bv) { bv = v; last = k; }
        }
        float* po = path_out + (size_t)(b0 + s) * T;
        po[T - 1] = (float)last;
        for (int t = T - 2; t >= 0; --t) {
            last = bps[t * KS + last];
            po[t] = (float)last;
        }
    }
}

// ---------------------------------------------------------------------------
// Fused launch: all independent forward work in one grid to fill the device.
// ---------------------------------------------------------------------------
__global__ __launch_bounds__(32)
void fused_crf_kernel(const float* __restrict__ seq_t, const float* __restrict__ tag_t,
                      const int* __restrict__ seq_y, const int* __restrict__ tag_y,
                      const float* __restrict__ tr_s, const float* __restrict__ tr_t,
                      float* __restrict__ seq_nll, float* __restrict__ tag_nll,
                      float* __restrict__ seq_path, float* __restrict__ tag_path)
{
    __shared__ __align__(16) char smem[SMEM_BYTES];
    int gid = blockIdx.x;
    if (gid < NB_TAG) { tag_nll_body(tag_t, tag_y, tr_t, tag_nll, gid * 16, smem); return; }
    gid -= NB_TAG;
    if (gid < NB_SEQ) { seq_nll_body(seq_t, seq_y, tr_s, seq_nll, gid * 32, smem); return; }
    gid -= NB_SEQ;
    if (gid < NB_VT)  { viterbi64_body(tag_t, tr_t, tag_path, gid, smem); return; }
    gid -= NB_VT;
    viterbi8_body(seq_t, tr_s, seq_path, gid * 4, smem);
}

// ---------------------------------------------------------------------------
// Uncertainty-weighted loss reduction (runs after fused kernel).
// ---------------------------------------------------------------------------
__global__ void loss_kernel(const float* __restrict__ seq_nll,
                            const float* __restrict__ tag_nll,
                            const float* __restrict__ log_vars,
                            float* __restrict__ out)
{
    __shared__ float s1[256], s2[256];
    const int tid = threadIdx.x;
    float a = 0.f, c = 0.f;
    for (int i = tid; i < B; i += 256) { a += seq_nll[i]; c += tag_nll[i]; }
    s1[tid] = a; s2[tid] = c;
    __syncthreads();
    for (int st = 128; st > 0; st >>= 1) {
        if (tid < st) { s1[tid] += s1[tid + st]; s2[tid] += s2[tid + st]; }
        __syncthreads();
    }
    if (tid == 0) {
        float lv0 = log_vars[0], lv1 = log_vars[1];
        out[0] = __expf(-lv0) * s1[0] / (float)B +
                 __expf(-lv1) * s2[0] / (float)B + lv0 + lv1;
    }
}

extern "C" void kernel_launch(void* const* d_in, const int* in_sizes, int n_in,
                              void* d_out, int out_size, void* d_ws, size_t ws_size,
                              hipStream_t stream) {
    (void)in_sizes; (void)n_in; (void)out_size; (void)ws_size;
    const float* seq_target = (const float*)d_in[0];
    const float* tag_target = (const float*)d_in[1];
    const int*   seq_true   = (const int*)d_in[2];
    const int*   tag_true   = (const int*)d_in[3];
    const float* trans_seq  = (const float*)d_in[4];
    const float* trans_tag  = (const float*)d_in[5];
    const float* log_vars   = (const float*)d_in[6];

    float* out      = (float*)d_out;
    float* seq_nll  = (float*)d_ws;
    float* tag_nll  = seq_nll + B;
    float* seq_path = out + 1;
    float* tag_path = out + 1 + (size_t)B * T;

    hipLaunchKernelGGL(fused_crf_kernel, dim3(NB_TAG + NB_SEQ + NB_VT + NB_VS), dim3(32), 0, stream,
                       seq_target, tag_target, seq_true, tag_true,
                       trans_seq, trans_tag, seq_nll, tag_nll, seq_path, tag_path);
    hipLaunchKernelGGL(loss_kernel, dim3(1), dim3(256), 0, stream,
                       seq_nll, tag_nll, log_vars, out);
}